// Geometrical_Pen_32822140076113
// MI455X (gfx1250) — compile-verified
//
#include <hip/hip_runtime.h>

typedef float v4f __attribute__((ext_vector_type(4)));

// One block of 256 threads (8 wave32) handles 1024 vertices.
// Per input array: 3072 floats = 12 KB; x-tile + y-tile = 24 KB LDS.
#define THREADS 256
#define VERTS_PER_BLOCK 1024
#define TILE_BYTES 12288      // per array
#define LDS_FLOATS 6144

__global__ __launch_bounds__(THREADS)
void geo_pen_main(const float* __restrict__ x, const float* __restrict__ y,
                  float* __restrict__ out) {
  __shared__ __align__(16) float lds[LDS_FLOATS]; // [0..3071]=x tile, [3072..6143]=y tile

  const unsigned t = threadIdx.x;
  const unsigned w = t >> 5;   // wave id within block
  const unsigned l = t & 31;   // lane id

  // Wave-relative LDS byte offset of the tile: low 32 bits of the flat address
  // of an LDS object are the LDS byte address (ISA 10.2 aperture mapping).
  const unsigned lds_base = (unsigned)(unsigned long long)&lds[0];

  const unsigned long long bf4 = (unsigned long long)blockIdx.x * 768ull; // float4s/array/block

  // Issue 6 async 128-bit copies into LDS (3 per array). Each wave fills only
  // the LDS region its own lanes read afterwards, so s_wait_asynccnt 0 is the
  // only synchronization needed (no workgroup barrier).
#pragma unroll
  for (int k = 0; k < 3; ++k) {
    const unsigned f4 = w * 96u + (unsigned)k * 32u + l; // float4 idx within tile
    const unsigned lx = lds_base + f4 * 16u;
    const unsigned ly = lx + TILE_BYTES;
    const unsigned long long gx = (unsigned long long)x + (bf4 + f4) * 16ull;
    const unsigned long long gy = (unsigned long long)y + (bf4 + f4) * 16ull;
    asm volatile("global_load_async_to_lds_b128 %0, %1, off"
                 :: "v"(lx), "v"(gx) : "memory");
    asm volatile("global_load_async_to_lds_b128 %0, %1, off"
                 :: "v"(ly), "v"(gy) : "memory");
  }

  // Uniform normalizer: L2 norm of row 0 of y only (quirk of the reference).
  const float y0 = y[0], y1 = y[1], y2 = y[2];
  const float inv = 1.0f / sqrtf(y0 * y0 + y1 * y1 + y2 * y2);

  asm volatile("s_wait_asynccnt 0x0" ::: "memory");

  // Lane l owns 4 vertices = 12 consecutive floats, 16B-aligned (48B stride).
  const float* lxp = &lds[w * 384u + l * 12u];
  const float* lyp = lxp + (TILE_BYTES / 4);
  const v4f xa = *(const v4f*)(lxp + 0);
  const v4f xb = *(const v4f*)(lxp + 4);
  const v4f xc = *(const v4f*)(lxp + 8);
  const v4f ya = *(const v4f*)(lyp + 0);
  const v4f yb = *(const v4f*)(lyp + 4);
  const v4f yc = *(const v4f*)(lyp + 8);

  v4f r;
  r.x = (xa.x * ya.x + xa.y * ya.y + xa.z * ya.z) * inv;
  r.y = (xa.w * ya.w + xb.x * yb.x + xb.y * yb.y) * inv;
  r.z = (xb.z * yb.z + xb.w * yb.w + xc.x * yc.x) * inv;
  r.w = (xc.y * yc.y + xc.z * yc.z + xc.w * yc.w) * inv;

  const unsigned long long vo =
      (unsigned long long)blockIdx.x * VERTS_PER_BLOCK + w * 128u + l * 4u;
  __builtin_nontemporal_store(r, (v4f*)(out + vo)); // streaming store, no reuse
}

// Scalar tail for any remainder (N = 2^24 is tile-divisible, but be safe).
__global__ void geo_pen_tail(const float* __restrict__ x, const float* __restrict__ y,
                             float* __restrict__ out, int start, int n) {
  const int i = start + (int)(blockIdx.x * blockDim.x + threadIdx.x);
  if (i < n) {
    const float y0 = y[0], y1 = y[1], y2 = y[2];
    const float inv = 1.0f / sqrtf(y0 * y0 + y1 * y1 + y2 * y2);
    out[i] = (x[3 * i] * y[3 * i] + x[3 * i + 1] * y[3 * i + 1] +
              x[3 * i + 2] * y[3 * i + 2]) * inv;
  }
}

extern "C" void kernel_launch(void* const* d_in, const int* in_sizes, int n_in,
                              void* d_out, int out_size, void* d_ws, size_t ws_size,
                              hipStream_t stream) {
  (void)in_sizes; (void)n_in; (void)d_ws; (void)ws_size;
  const float* x = (const float*)d_in[0];
  const float* y = (const float*)d_in[1];
  float* out = (float*)d_out;
  const int n = out_size; // number of vertices
  const int nb = n / VERTS_PER_BLOCK;
  if (nb > 0) {
    geo_pen_main<<<dim3(nb), dim3(THREADS), 0, stream>>>(x, y, out);
  }
  const int done = nb * VERTS_PER_BLOCK;
  const int tail = n - done;
  if (tail > 0) {
    geo_pen_tail<<<dim3((tail + 255) / 256), dim3(256), 0, stream>>>(x, y, out, done, n);
  }
}